// AMTCL_66520453480770
// MI455X (gfx1250) — compile-verified
//
#include <hip/hip_runtime.h>
#include <hip/hip_bf16.h>
#include <math.h>

typedef __attribute__((ext_vector_type(2))) float v2f;
typedef __attribute__((ext_vector_type(8))) float v8f;

#define B_N   4096
#define C_N   100
#define D_N   384
#define CPAD  112          // 7 N-tiles of 16
#define KTOT  768          // [inputs^2 ; inputs] stacked along K
#define EPSV  1e-12f

// workspace layout (in floats)
#define WG_OFF   0
#define WG_FLTS  ((KTOT/2)*CPAD*2)        // 86016 : pair-packed W matrix (k even/odd interleaved)
#define A_OFF    (WG_OFF + WG_FLTS)       // 112   : A[c] = sum_d w2*centers^2
#define CD_OFF   (A_OFF + CPAD)           // 112   : centers_dist[c]
#define DVAL_OFF (CD_OFF + CPAD)          // 4096*112 : dval[i][c]
// total = 544,992 floats = ~2.08 MB

// ---------------------------------------------------------------------------
// Kernel 1: build A[c] and the pair-packed GEMM B-operand W.
//   W[k][c]: k<384 -> w2[c,k] ; k>=384 -> -2*w2[c,k-384]*centers[c,k-384]
//   packed as Wg2[p*CPAD + c] = { W[2p][c], W[2p+1][c] }
__global__ void k1_prep(const float* __restrict__ centers,
                        const float* __restrict__ cw,
                        float* __restrict__ ws) {
    __shared__ float red[128];
    const int c   = blockIdx.x;          // 0..111
    const int tid = threadIdx.x;         // 128 threads
    float* Wg   = ws + WG_OFF;
    float* Aarr = ws + A_OFF;
    float acc = 0.f;
    if (c < C_N) {
        for (int d = tid; d < D_N; d += 128) {
            float w2v = exp2f(cw[c*D_N + d]);
            float cv  = centers[c*D_N + d];
            int p = d >> 1, s = d & 1;
            Wg[(p*CPAD + c)*2 + s]       = w2v;               // quad half (k=d)
            Wg[((p+192)*CPAD + c)*2 + s] = -2.0f*w2v*cv;      // linear half (k=d+384)
            acc += w2v*cv*cv;
        }
    } else {
        for (int d = tid; d < D_N; d += 128) {
            int p = d >> 1, s = d & 1;
            Wg[(p*CPAD + c)*2 + s]       = 0.f;
            Wg[((p+192)*CPAD + c)*2 + s] = 0.f;
        }
    }
    red[tid] = acc;
    __syncthreads();
    for (int off = 64; off > 0; off >>= 1) {
        if (tid < off) red[tid] += red[tid + off];
        __syncthreads();
    }
    if (tid == 0) Aarr[c] = (c < C_N) ? red[0] : 0.f;
}

// ---------------------------------------------------------------------------
// Kernel 2: centers_dist[c] = min_{j!=c} sqrt(max(0, sum_d w2[c,d]*(centers[c,d]-centers[j,d])^2))
__global__ void k2_cdist(const float* __restrict__ centers,
                         const float* __restrict__ cw,
                         float* __restrict__ ws) {
    __shared__ float w2s[D_N];
    __shared__ float cs[D_N];
    __shared__ float wmin[4];
    const int c   = blockIdx.x;          // 0..99
    const int tid = threadIdx.x;         // 128 threads = 4 waves
    for (int d = tid; d < D_N; d += 128) {
        w2s[d] = exp2f(cw[c*D_N + d]);
        cs[d]  = centers[c*D_N + d];
    }
    __syncthreads();
    const int wave = tid >> 5, lane = tid & 31;
    float mn = INFINITY;
    for (int j = wave; j < C_N; j += 4) {
        if (j == c) continue;            // uniform per wave
        float p = 0.f;
        for (int d = lane; d < D_N; d += 32) {
            float diff = cs[d] - centers[j*D_N + d];
            p += w2s[d]*diff*diff;
        }
        for (int off = 16; off > 0; off >>= 1) p += __shfl_down(p, off, 32);
        if (lane == 0) mn = fminf(mn, p);
    }
    if (lane == 0) wmin[wave] = mn;
    __syncthreads();
    if (tid == 0) {
        float m = fminf(fminf(wmin[0], wmin[1]), fminf(wmin[2], wmin[3]));
        ws[CD_OFF + c] = sqrtf(fmaxf(m, 0.f));
    }
}

// ---------------------------------------------------------------------------
// Kernel 3: fp32 WMMA GEMM. dval[i][c] = A[c] + X[i,:] @ W[:,c], X = [inputs^2 ; inputs].
// Block = 128 thr (4 waves); block tile = 256 rows x 16 cols.
// Each wave owns FOUR 16x16 sub-tiles (64 rows): 4 independent accumulator
// chains per k-step, and each B fragment fetched from LDS feeds 4 WMMAs.
__global__ void k3_gemm(const float* __restrict__ inputs,
                        float* __restrict__ ws) {
    __shared__ float2 Bs[(KTOT/2)*16];   // 6144 float2 = 48 KB, pair layout [p][n]
    const int tid = threadIdx.x;
    const int by  = blockIdx.y;          // N tile (0..6)
    const float2* Wg2 = (const float2*)(ws + WG_OFF);
    // stage B tile (uniform trip count, EXEC stays full for the WMMA loop)
    for (int q = tid; q < (KTOT/2)*16; q += 128) {
        int p = q >> 4, n = q & 15;
        Bs[q] = Wg2[p*CPAD + by*16 + n];
    }
    __syncthreads();

    const int wave = tid >> 5, lane = tid & 31;
    const int half = lane >> 4, n = lane & 15;
    const int rbase = blockIdx.x*256 + wave*64;        // wave's 64-row strip
    const float2* in2 = (const float2*)inputs;
    // A-frag: M = lane&15 in both halves; half selects K{0,1} vs K{2,3}
    int rowb[4];
    #pragma unroll
    for (int s = 0; s < 4; s++) rowb[s] = (rbase + s*16 + n) * (D_N/2);

    v8f acc0 = {}, acc1 = {}, acc2 = {}, acc3 = {};
    #pragma unroll 2
    for (int kk = 0; kk < D_N; kk += 4) {
        const int pq = (kk >> 1) + half;
        float2 bq = Bs[pq*16 + n];                     // quad-half W rows (k=kk..kk+3)
        float2 bl = Bs[(192 + pq)*16 + n];             // linear-half W rows
        v2f b1; b1.x = bq.x; b1.y = bq.y;
        v2f b2; b2.x = bl.x; b2.y = bl.y;

        float2 x0 = in2[rowb[0] + pq];
        float2 x1 = in2[rowb[1] + pq];
        float2 x2 = in2[rowb[2] + pq];
        float2 x3 = in2[rowb[3] + pq];

        v2f q0; q0.x = x0.x*x0.x; q0.y = x0.y*x0.y;
        v2f q1; q1.x = x1.x*x1.x; q1.y = x1.y*x1.y;
        v2f q2; q2.x = x2.x*x2.x; q2.y = x2.y*x2.y;
        v2f q3; q3.x = x3.x*x3.x; q3.y = x3.y*x3.y;
        v2f l0; l0.x = x0.x;      l0.y = x0.y;
        v2f l1; l1.x = x1.x;      l1.y = x1.y;
        v2f l2; l2.x = x2.x;      l2.y = x2.y;
        v2f l3; l3.x = x3.x;      l3.y = x3.y;

        // interleaved: 4 independent chains, dependent WMMAs >= 4 apart
        acc0 = __builtin_amdgcn_wmma_f32_16x16x4_f32(false, q0, false, b1, (short)0, acc0, false, false);
        acc1 = __builtin_amdgcn_wmma_f32_16x16x4_f32(false, q1, false, b1, (short)0, acc1, false, false);
        acc2 = __builtin_amdgcn_wmma_f32_16x16x4_f32(false, q2, false, b1, (short)0, acc2, false, false);
        acc3 = __builtin_amdgcn_wmma_f32_16x16x4_f32(false, q3, false, b1, (short)0, acc3, false, false);
        acc0 = __builtin_amdgcn_wmma_f32_16x16x4_f32(false, l0, false, b2, (short)0, acc0, false, false);
        acc1 = __builtin_amdgcn_wmma_f32_16x16x4_f32(false, l1, false, b2, (short)0, acc1, false, false);
        acc2 = __builtin_amdgcn_wmma_f32_16x16x4_f32(false, l2, false, b2, (short)0, acc2, false, false);
        acc3 = __builtin_amdgcn_wmma_f32_16x16x4_f32(false, l3, false, b2, (short)0, acc3, false, false);
    }

    const int col = by*16 + n;
    const float Ac = ws[A_OFF + col];
    float* dval = ws + DVAL_OFF;
    // D-frag: VGPR r holds M = r (lanes 0-15) / r+8 (lanes 16-31)
    #pragma unroll
    for (int s = 0; s < 4; s++) {
        v8f a = (s == 0) ? acc0 : (s == 1) ? acc1 : (s == 2) ? acc2 : acc3;
        const int mbase = rbase + s*16 + half*8;
        #pragma unroll
        for (int r = 0; r < 8; r++) {
            dval[(mbase + r)*CPAD + col] = a[r] + Ac;
        }
    }
}

// ---------------------------------------------------------------------------
// Kernel 4: class counts, hardest pos/neg per row, final loss scalar.
__global__ void k4_final(const float* __restrict__ ws,
                         const int*   __restrict__ targets,
                         float* __restrict__ out) {
    __shared__ int   counts[CPAD];
    __shared__ float redw[32];
    const int tid = threadIdx.x;         // 1024 threads = 32 waves
    if (tid < CPAD) counts[tid] = 0;
    __syncthreads();
    for (int i = tid; i < B_N; i += 1024) atomicAdd(&counts[targets[i]], 1);
    __syncthreads();

    const float* dval  = ws + DVAL_OFF;
    const float* cdist = ws + CD_OFF;
    float lsum = 0.f;
    for (int i = tid; i < B_N; i += 1024) {
        int t = targets[i];
        const float* row = dval + i*CPAD;
        float dpos2 = row[t];
        float dneg2 = INFINITY;
        for (int c = 0; c < C_N; c++) {
            float v = row[c];
            bool valid = (c != t) && (counts[c] > 0);
            dneg2 = valid ? fminf(dneg2, v) : dneg2;
        }
        float d_pos = sqrtf(fmaxf(dpos2, EPSV));
        float d_neg = sqrtf(fmaxf(dneg2, EPSV));   // INF stays INF
        float cdv = cdist[t];
        lsum += (d_neg >= cdv) ? d_pos : (d_pos - d_neg + cdv);
    }
    for (int off = 16; off > 0; off >>= 1) lsum += __shfl_down(lsum, off, 32);
    const int wave = tid >> 5, lane = tid & 31;
    if (lane == 0) redw[wave] = lsum;
    __syncthreads();
    if (wave == 0) {
        float v = redw[lane];
        for (int off = 16; off > 0; off >>= 1) v += __shfl_down(v, off, 32);
        if (lane == 0) out[0] = v * (1.0f / (float)B_N);
    }
}

// ---------------------------------------------------------------------------
extern "C" void kernel_launch(void* const* d_in, const int* in_sizes, int n_in,
                              void* d_out, int out_size, void* d_ws, size_t ws_size,
                              hipStream_t stream) {
    const float* inputs  = (const float*)d_in[0];   // [4096,384]
    const float* centers = (const float*)d_in[1];   // [100,384]
    const float* cw      = (const float*)d_in[2];   // [100,384]
    const int*   targets = (const int*)d_in[3];     // [4096]
    // d_in[4] = epoch_number (unused by the math)
    float* ws  = (float*)d_ws;
    float* out = (float*)d_out;

    k1_prep <<<dim3(CPAD),             dim3(128),  0, stream>>>(centers, cw, ws);
    k2_cdist<<<dim3(C_N),              dim3(128),  0, stream>>>(centers, cw, ws);
    k3_gemm <<<dim3(B_N/256, CPAD/16), dim3(128),  0, stream>>>(inputs, ws);
    k4_final<<<dim3(1),                dim3(1024), 0, stream>>>(ws, targets, out);
}